// Diffusion_DEM_82557861364401
// MI455X (gfx1250) — compile-verified
//
#include <hip/hip_runtime.h>
#include <math.h>

typedef __attribute__((ext_vector_type(2))) float v2f;
typedef __attribute__((ext_vector_type(8))) float v8f;
typedef __attribute__((ext_vector_type(4))) unsigned v4u;
typedef __attribute__((ext_vector_type(8))) int v8i;
typedef __attribute__((ext_vector_type(4))) int v4i;

#if defined(__has_builtin)
#if __has_builtin(__builtin_amdgcn_tensor_load_to_lds) && \
    __has_builtin(__builtin_amdgcn_s_wait_tensorcnt)
#define HAVE_TDM 1
#endif
#endif
#ifndef HAVE_TDM
#define HAVE_TDM 0
#endif

// ---- physical constants (folded at compile time, matching the reference) ----
constexpr float ALFA  = (float)(0.001 * 8e-07 * 3.497e-06 / 7.08e-15);
constexpr float THETA = (float)(3.497e-06 * 1e10 / (8.3145 * 300.0 * 3.0 * 0.4));
constexpr float LAM1  = (float)(1.0 / (1.0 - 0.6));
constexpr float LAM2  = (float)(0.7 / 1.3);
constexpr float LAM3  = (float)(0.3 / 1.3);
constexpr float FSC   = (float)(1.3 / (0.7 * 3.0)); // multiplies ALFA in f_stress

// workspace layout (floats): [0..6] loss accumulators,
// uint slots [8]=min_x [9]=min_t [10]=max_x [11]=max_t (bit-ordered, data >= 0)

__global__ void pinn_init_ws(float* wsf) {
  int t = threadIdx.x;
  if (t < 8) wsf[t] = 0.f;
  unsigned* u = (unsigned*)wsf;
  if (t == 8 || t == 9)  u[t] = 0x7f800000u; // +inf bits
  if (t == 10 || t == 11) u[t] = 0u;
}

__global__ void pinn_minmax(const float* __restrict__ X, int n, float* wsf) {
  unsigned* mm = (unsigned*)wsf;
  float mnx = __builtin_inff(), mnt = __builtin_inff();
  float mxx = 0.f, mxt = 0.f; // data is non-negative
  for (int i = blockIdx.x * blockDim.x + threadIdx.x; i < n;
       i += gridDim.x * blockDim.x) {
    float x = X[2 * i], t = X[2 * i + 1];
    mnx = fminf(mnx, x); mxx = fmaxf(mxx, x);
    mnt = fminf(mnt, t); mxt = fmaxf(mxt, t);
  }
  atomicMin(&mm[8],  __float_as_uint(mnx));
  atomicMin(&mm[9],  __float_as_uint(mnt));
  atomicMax(&mm[10], __float_as_uint(mxx));
  atomicMax(&mm[11], __float_as_uint(mxt));
}

// jets: per point 5 components (val, d/dx, d2/dx2, d3/dx3, d/dt)
// LDS: two ping-pong activation buffers of 5 planes x [16 pts x 128 neurons],
// a jets scratch area, and a 64 KB TDM-staged weight buffer (one layer).
#define HB_FLOATS  (5 * 2048)
#define JETS_OFF   (2 * HB_FLOATS)
#define WST_OFF    (2 * HB_FLOATS + 160)
#define SMEM_FLOATS (WST_OFF + 128 * 128)

__global__ __launch_bounds__(128) void pinn_net(
    const float* __restrict__ X, int npts,
    const float* __restrict__ W0, const float* __restrict__ b0,
    const float* __restrict__ W1, const float* __restrict__ b1,
    const float* __restrict__ W2, const float* __restrict__ b2,
    const float* __restrict__ W3, const float* __restrict__ b3,
    float* wsf, const float* __restrict__ Yin, const float* __restrict__ Win,
    int mode) {
  extern __shared__ float smem[];
  float* buf0 = smem;
  float* buf1 = smem + HB_FLOATS;
  float* jets = smem + JETS_OFF;  // [16 pts][5 comps][2 outs]
  float* wst  = smem + WST_OFF;   // TDM-staged 128x128 weight matrix

  const int tid  = threadIdx.x;
  const int lane = tid & 31;
  const int wid  = tid >> 5;
  const int lpt  = lane & 15;         // M index (point within tile)
  const int kh   = (lane >> 4) << 1;  // K offset 0 or 2 (A/B layout halves)
  const int prow = (lane >> 4) << 3;  // C/D: M = v (+8 for upper lanes)
  const int pt0  = blockIdx.x * 16;

  const unsigned* mm = (const unsigned*)wsf;
  const float lbx = __uint_as_float(mm[8]);
  const float lbt = __uint_as_float(mm[9]);
  const float ubx = __uint_as_float(mm[10]);
  const float ubt = __uint_as_float(mm[11]);
  const float gx = 2.f / (ubx - lbx); // d(xhat)/dx
  const float gt = 2.f / (ubt - lbt); // d(that)/dt

  // ---------------- layer 0: 2 -> 128, seed jets, tanh ----------------
  for (int e = tid; e < 2048; e += 128) {
    int pt = e >> 7, n = e & 127;
    float xr = X[(pt0 + pt) * 2 + 0];
    float tr = X[(pt0 + pt) * 2 + 1];
    float xh = (xr - lbx) * gx - 1.f;
    float th = (tr - lbt) * gt - 1.f;
    float w0n = W0[n], w1n = W0[128 + n];
    float z0 = fmaf(xh, w0n, fmaf(th, w1n, b0[n]));
    float z1 = gx * w0n; // dz/dx   (z2 = z3 = 0: input linear in x)
    float zt = gt * w1n; // dz/dt
    float y0 = tanhf(z0);
    float s = 1.f - y0 * y0;            // tanh'
    float fpp = -2.f * y0 * s;          // tanh''
    float fppp = 4.f * y0 * y0 * s - 2.f * s * s; // tanh'''
    buf0[e]            = y0;
    buf0[2048 + e]     = s * z1;
    buf0[2 * 2048 + e] = fpp * z1 * z1;
    buf0[3 * 2048 + e] = fppp * z1 * z1 * z1;
    buf0[4 * 2048 + e] = s * zt;
  }

  // ---------------- hidden layers 128 -> 128 via fp32 WMMA ----------------
  for (int L = 0; L < 2; ++L) {
    const float* Wl = L ? W2 : W1;
    const float* bl = L ? b2 : b1;
    float* cur = L ? buf1 : buf0;
    float* nxt = L ? buf0 : buf1;

    // Stage this layer's 128x128 weights into LDS with the Tensor Data Mover.
    // D# groups per CDNA5 ISA ch.8: 2-D tensor, tile == full matrix, f32.
#if HAVE_TDM
    if (wid == 0) {
      unsigned long long ga = (unsigned long long)(uintptr_t)Wl;
      v4u g0;
      g0[0] = 1u;                                   // count=1, user descriptor
      g0[1] = (unsigned)(WST_OFF * sizeof(float));  // lds_addr (bytes)
      g0[2] = (unsigned)(ga & 0xFFFFFFFFu);         // global_addr[31:0]
      g0[3] = (unsigned)((ga >> 32) & 0x01FFFFFFu) | (2u << 30); // addr|type=2
      v8i g1;
      g1[0] = (int)(2u << 16);       // workgroup_mask=0, data_size=4B
      g1[1] = (int)(128u << 16);     // tensor_dim0[15:0]=128 (no atomic bar)
      g1[2] = (int)(128u << 16);     // tensor_dim0 hi=0, tensor_dim1 lo=128
      g1[3] = (int)(128u << 16);     // tensor_dim1 hi=0, tile_dim0=128
      g1[4] = 128;                   // tile_dim1=128, tile_dim2=0
      g1[5] = 128;                   // tensor_dim0_stride[31:0]=128
      g1[6] = (int)(16384u << 16);   // stride0 hi=0, tensor_dim1_stride lo
      g1[7] = 0;
      v4i gz = (v4i)0;
#if __clang_major__ >= 23
      __builtin_amdgcn_tensor_load_to_lds(g0, g1, gz, gz, (v8i)0, 0);
#else
      __builtin_amdgcn_tensor_load_to_lds(g0, g1, gz, gz, 0);
#endif
      __builtin_amdgcn_s_wait_tensorcnt(0);
    }
#else
    for (int e = tid; e < 128 * 128; e += 128) wst[e] = Wl[e];
#endif
    __syncthreads(); // layer-0/tanh results + staged weights visible

    for (int nt = wid; nt < 8; nt += 4) { // each wave: 2 output tiles of 16
      const int ncol = nt * 16 + lpt;
      v2f breg[32]; // B-layout weight tile for full K=128, reused for 5 comps
      #pragma unroll
      for (int k = 0; k < 32; ++k) {
        breg[k][0] = wst[(4 * k + kh) * 128 + ncol];
        breg[k][1] = wst[(4 * k + kh + 1) * 128 + ncol];
      }
      #pragma unroll
      for (int c = 0; c < 5; ++c) {
        const float* hc = cur + c * 2048 + lpt * 128 + kh;
        v8f acc = {0.f, 0.f, 0.f, 0.f, 0.f, 0.f, 0.f, 0.f};
        #pragma unroll
        for (int k = 0; k < 32; ++k) {
          v2f a;
          a[0] = hc[4 * k];
          a[1] = hc[4 * k + 1];
          acc = __builtin_amdgcn_wmma_f32_16x16x4_f32(
              false, a, false, breg[k], (short)0, acc, false, false);
        }
        float* hn = nxt + c * 2048 + nt * 16 + lpt;
        #pragma unroll
        for (int v = 0; v < 8; ++v) hn[(v + prow) * 128] = acc[v];
      }
    }
    __syncthreads(); // all reads of cur/wst and writes of nxt complete

    // tanh jet mixing (Faa di Bruno), bias folded into component 0
    for (int e = tid; e < 2048; e += 128) {
      int n = e & 127;
      float z0 = nxt[e] + bl[n];
      float z1 = nxt[2048 + e];
      float z2 = nxt[2 * 2048 + e];
      float z3 = nxt[3 * 2048 + e];
      float zt = nxt[4 * 2048 + e];
      float y0 = tanhf(z0);
      float s = 1.f - y0 * y0;
      float fpp = -2.f * y0 * s;
      float fppp = 4.f * y0 * y0 * s - 2.f * s * s;
      nxt[e]            = y0;
      nxt[2048 + e]     = s * z1;
      nxt[2 * 2048 + e] = fmaf(fpp, z1 * z1, s * z2);
      nxt[3 * 2048 + e] = s * z3 + 3.f * fpp * z1 * z2 + fppp * z1 * z1 * z1;
      nxt[4 * 2048 + e] = s * zt;
    }
    if (L == 1) __syncthreads(); // L==0 staging barrier covers the other case
  }

  // ---------------- output layer 128 -> 2 (jets in buf0) ----------------
  for (int d = tid; d < 160; d += 128) {
    int out = d % 2;
    int comp = (d / 2) % 5;
    int pt = d / 10;
    const float* h = buf0 + comp * 2048 + pt * 128;
    float sacc = 0.f;
    for (int n = 0; n < 128; ++n) sacc = fmaf(h[n], W3[2 * n + out], sacc);
    if (comp == 0) sacc += b3[out];
    jets[d] = sacc;
  }
  __syncthreads();

  // ---------------- residuals + loss accumulation ----------------
  if (tid < 16) {
    int pt = tid;
    const float* J = jets + pt * 10; // [comp][out]
    float u = J[0],  c = J[1];
    float ux = J[2], cx = J[3];
    float uxx = J[4], cxx = J[5];
    float uxxx = J[6];
    float ct = J[9];
    float x = X[(pt0 + pt) * 2];
    if (mode == 0) { // interior
      float x2 = x * x, x3 = x2 * x;
      float fs = x2 * uxx + x * ux - u - x2 * cx * (ALFA * FSC);
      float inner1 = 2.f * x2 * uxx - x * ux + u - ALFA * x2 * cx +
                     x3 * uxxx - ALFA * x3 * cxx;
      float inner2 = x2 * uxx + x * ux - u - ALFA * cx * x2;
      float resid = x3 * ct - x3 * cxx - x2 * cx + THETA * c * inner1 +
                    THETA * x * cx * inner2;
      atomicAdd(&wsf[0], fs * fs);
      atomicAdd(&wsf[1], resid * resid);
    } else if (mode == 1 || mode == 2) { // boundary (rhs 0 / 1)
      float rhs = (mode == 2) ? 1.f : 0.f;
      float x2 = x * x;
      float stress = LAM1 * (LAM2 * ux + LAM3 * u / x - ALFA * c / 3.f);
      float fd = x2 * cx - THETA * c * (x2 * uxx + x * ux - u - ALFA * cx * x2) -
                 rhs * x2;
      int sA = (mode == 1) ? 2 : 4;
      atomicAdd(&wsf[sA],     stress * stress);
      atomicAdd(&wsf[sA + 1], fd * fd);
    } else { // init
      int gi = (pt0 + pt) * 2;
      float acc2 = 0.f;
      for (int j = 0; j < 2; ++j) {
        float du = u - Yin[gi + j];
        float dc = c - Yin[gi + j];
        acc2 += (du * du + dc * dc) * Win[gi + j];
      }
      atomicAdd(&wsf[6], acc2);
    }
  }
}

__global__ void pinn_finalize(const float* wsf, float* out, float Ni, float Nb,
                              float Nit2) {
  if (threadIdx.x == 0 && blockIdx.x == 0)
    out[0] = (wsf[0] + wsf[1]) / Ni +
             (wsf[2] + wsf[3] + wsf[4] + wsf[5]) / Nb + wsf[6] / Nit2;
}

extern "C" void kernel_launch(void* const* d_in, const int* in_sizes, int n_in,
                              void* d_out, int out_size, void* d_ws,
                              size_t ws_size, hipStream_t stream) {
  const float* Xint  = (const float*)d_in[0];
  const float* Xbl   = (const float*)d_in[3];
  const float* Xinit = (const float*)d_in[6];
  const float* Yinit = (const float*)d_in[7];
  const float* Winit = (const float*)d_in[8];
  const float* Xbr   = (const float*)d_in[9];
  const float* W0 = (const float*)d_in[11];
  const float* b0 = (const float*)d_in[12];
  const float* W1 = (const float*)d_in[13];
  const float* b1 = (const float*)d_in[14];
  const float* W2 = (const float*)d_in[15];
  const float* b2 = (const float*)d_in[16];
  const float* W3 = (const float*)d_in[17];
  const float* b3 = (const float*)d_in[18];

  float* wsf = (float*)d_ws;
  float* out = (float*)d_out;

  const int Ni  = in_sizes[0] / 2; // 65536
  const int Nb  = in_sizes[3] / 2; // 8192
  const int Nit = in_sizes[6] / 2; // 8192
  const size_t smem = SMEM_FLOATS * sizeof(float);

  pinn_init_ws<<<1, 32, 0, stream>>>(wsf);
  pinn_minmax<<<64, 256, 0, stream>>>(Xint, Ni, wsf);

  pinn_net<<<Ni / 16, 128, smem, stream>>>(Xint, Ni, W0, b0, W1, b1, W2, b2,
                                           W3, b3, wsf, Yinit, Winit, 0);
  pinn_net<<<Nb / 16, 128, smem, stream>>>(Xbl, Nb, W0, b0, W1, b1, W2, b2,
                                           W3, b3, wsf, Yinit, Winit, 1);
  pinn_net<<<Nb / 16, 128, smem, stream>>>(Xbr, Nb, W0, b0, W1, b1, W2, b2,
                                           W3, b3, wsf, Yinit, Winit, 2);
  pinn_net<<<Nit / 16, 128, smem, stream>>>(Xinit, Nit, W0, b0, W1, b1, W2, b2,
                                            W3, b3, wsf, Yinit, Winit, 3);

  pinn_finalize<<<1, 1, 0, stream>>>(wsf, out, (float)Ni, (float)Nb,
                                     (float)(2 * Nit));
}